// TaskAlignedAssigner_46909632807050
// MI455X (gfx1250) — compile-verified
//
#include <hip/hip_runtime.h>
#include <hip/hip_bf16.h>
#include <math.h>

// ---------------------------------------------------------------------------
// TaskAlignedAssigner for MI455X (gfx1250, wave32).
// Bandwidth-bound problem (~190MB traffic @ 23.3TB/s). The scattered
// score-gather pd_scores[b,a,label[b,m]] is reformulated as a one-hot GEMM
// executed on the matrix pipe with V_WMMA_F32_16X16X4_F32 (exact arithmetic).
// Score tiles are staged global->LDS with GLOBAL_LOAD_ASYNC_TO_LDS_B128
// (ASYNCcnt-tracked DMA, no VGPR round-trip), then consumed as WMMA B-frags.
// ---------------------------------------------------------------------------

#define B_    32
#define A_    8400
#define M_    32
#define NC_   80
#define TOPK_ 13
#define EPS_  1e-9f
#define CEPS_ 1e-7f

typedef float v2f __attribute__((ext_vector_type(2)));
typedef float v8f __attribute__((ext_vector_type(8)));

// ---------------------------------------------------------------------------
// Kernel 1: cls[b,m,a] = pd_scores[b,a,labels[b,m]] as one-hot GEMM.
// One wave per (b, 16-wide a-tile). Two f32 accumulators cover M=32.
// A-frag (16x4 f32): lanes 0-15 -> M=lane, K=v;   lanes 16-31 -> K=v+2.
// B-frag (4x16 f32): lanes 0-15 -> N=lane, K=v;   lanes 16-31 -> K=v+2.
// D (16x16 f32): VGPR r -> M=r (lanes 0-15) / M=r+8 (lanes 16-31), N=lane%16.
// ---------------------------------------------------------------------------
__global__ __launch_bounds__(32) void k_cls_wmma(const float* __restrict__ scores,
                                                 const int*   __restrict__ labels,
                                                 float*       __restrict__ cls) {
  const int b    = blockIdx.y;
  const int a0   = blockIdx.x * 16;
  const int lane = threadIdx.x;

  // 16 anchors x 80 classes (+pad). Row stride 336B (multiple of 16).
  // Sole static-LDS object -> starts at LDS offset 0.
  __shared__ __attribute__((aligned(16))) float lds[16][NC_ + 4];

  // Stage the 16x80 f32 tile straight into LDS via async DMA (B128/lane).
  const float* src = scores + ((size_t)b * A_ + a0) * NC_;
  for (int it = 0; it < 10; ++it) {
    const int idx4 = it * 32 + lane;          // 0..319 (20 float4 per row)
    const int row  = idx4 / 20;
    const int c4   = (idx4 % 20) * 4;
    const unsigned ldsOff = (unsigned)((row * (NC_ + 4) + c4) * sizeof(float));
    const unsigned long long gaddr =
        (unsigned long long)(src + (size_t)row * NC_ + c4);
    asm volatile("global_load_async_to_lds_b128 %0, %1, off"
                 :: "v"(ldsOff), "v"(gaddr) : "memory");
  }

  // Overlap: set up one-hot label registers while the DMA is in flight.
  const int nrow = lane & 15;          // N index (anchor within tile) / M row
  const int hi   = lane >> 4;          // lane group selects K sub-pair
  const int lab0 = labels[b * M_ + nrow];        // one-hot row, m-block 0
  const int lab1 = labels[b * M_ + nrow + 16];   // one-hot row, m-block 1

  asm volatile("s_wait_asynccnt 0x0" ::: "memory");
  __syncthreads();

  v8f acc0 = {};   // D for m = 0..15
  v8f acc1 = {};   // D for m = 16..31
  for (int k0 = 0; k0 < NC_; k0 += 4) {
    const int cb = k0 + hi * 2;
    v2f bf; bf.x = lds[nrow][cb]; bf.y = lds[nrow][cb + 1];
    v2f a0f, a1f;
    a0f.x = (lab0 == cb    ) ? 1.0f : 0.0f;
    a0f.y = (lab0 == cb + 1) ? 1.0f : 0.0f;
    a1f.x = (lab1 == cb    ) ? 1.0f : 0.0f;
    a1f.y = (lab1 == cb + 1) ? 1.0f : 0.0f;
    acc0 = __builtin_amdgcn_wmma_f32_16x16x4_f32(false, a0f, false, bf,
                                                 (short)0, acc0, false, false);
    acc1 = __builtin_amdgcn_wmma_f32_16x16x4_f32(false, a1f, false, bf,
                                                 (short)0, acc1, false, false);
  }

  const int mBase = hi * 8;
  #pragma unroll
  for (int r = 0; r < 8; ++r) {
    const int m = mBase + r;
    cls[((size_t)(b * M_ + m     )) * A_ + a0 + nrow] = acc0[r];
    cls[((size_t)(b * M_ + m + 16)) * A_ + a0 + nrow] = acc1[r];
  }
}

// ---------------------------------------------------------------------------
// Kernel 2: per (b,a) thread, loop over m: mask_in_gt, CIoU, align.
// arr0: in = cls, out = mask_in_gt (same element, read-then-write per thread).
// ---------------------------------------------------------------------------
__global__ __launch_bounds__(256) void k_pair(const float* __restrict__ pd_bboxes,
                                              const float* __restrict__ anc,
                                              const float* __restrict__ gt_bboxes,
                                              const float* __restrict__ gt_mask,
                                              float* arr0,                 // cls -> mask_in_gt
                                              float* __restrict__ align_o, // align
                                              float* __restrict__ iou_o) { // iou
  const int b   = blockIdx.y;
  const int tid = threadIdx.x;
  const int a   = blockIdx.x * 256 + tid;

  __shared__ float gx1[M_], gy1[M_], gx2[M_], gy2[M_];
  __shared__ float gmv[M_], gat[M_], gar[M_];
  if (tid < M_) {
    const float4 gb = ((const float4*)gt_bboxes)[b * M_ + tid];
    gx1[tid] = gb.x; gy1[tid] = gb.y; gx2[tid] = gb.z; gy2[tid] = gb.w;
    gmv[tid] = gt_mask[b * M_ + tid];
    const float w1 = gb.z - gb.x;
    const float h1 = gb.w - gb.y + CEPS_;
    gat[tid] = atanf(w1 / h1);
    gar[tid] = w1 * h1;
  }
  __syncthreads();
  if (a >= A_) return;

  const float4 pb = ((const float4*)pd_bboxes)[(size_t)b * A_ + a];
  const float2 xy = ((const float2*)anc)[a];
  const float w2    = pb.z - pb.x;
  const float h2    = pb.w - pb.y + CEPS_;
  const float at2   = atanf(w2 / h2);
  const float area2 = w2 * h2;
  const float inv_pi2 = 4.0f / (float)(M_PI * M_PI);

  for (int m = 0; m < M_; ++m) {
    const size_t off = ((size_t)(b * M_ + m)) * A_ + a;
    const float d0 = xy.x - gx1[m], d1 = xy.y - gy1[m];
    const float d2 = gx2[m] - xy.x, d3 = gy2[m] - xy.y;
    const float mind  = fminf(fminf(d0, d1), fminf(d2, d3));
    const float in_gt = (mind > EPS_) ? 1.0f : 0.0f;
    const bool  fine  = (in_gt * gmv[m]) > 0.0f;

    float clsv = 0.0f, iouv = 0.0f;
    if (fine) {
      clsv = arr0[off];
      const float ix = fmaxf(fminf(gx2[m], pb.z) - fmaxf(gx1[m], pb.x), 0.0f);
      const float iy = fmaxf(fminf(gy2[m], pb.w) - fmaxf(gy1[m], pb.y), 0.0f);
      const float inter = ix * iy;
      const float uni   = gar[m] + area2 - inter + CEPS_;
      const float iou   = inter / uni;
      const float cw = fmaxf(gx2[m], pb.z) - fminf(gx1[m], pb.x);
      const float ch = fmaxf(gy2[m], pb.w) - fminf(gy1[m], pb.y);
      const float c2 = cw * cw + ch * ch + CEPS_;
      const float dx = pb.x + pb.z - gx1[m] - gx2[m];
      const float dy = pb.y + pb.w - gy1[m] - gy2[m];
      const float rho2  = (dx * dx + dy * dy) * 0.25f;
      const float dat   = at2 - gat[m];
      const float v     = inv_pi2 * dat * dat;
      const float alpha = v / (v - iou + 1.0f + CEPS_);
      iouv = fmaxf(iou - (rho2 / c2 + v * alpha), 0.0f);
    }
    const float p2 = iouv * iouv;
    align_o[off] = clsv * (p2 * p2 * p2);   // cls^1 * iou^6
    iou_o[off]   = iouv;
    arr0[off]    = in_gt;                    // repurpose arr0 as mask_in_gt
  }
}

// ---------------------------------------------------------------------------
// Kernel 3: per (b,m) top-13 of align with jax.lax.top_k tie semantics
// (strict lexicographic (value desc, index asc)). Invalid GT rows -> zeros
// (reference maps them all to index 0 and kills them via counts>1).
// maskbuf: in = mask_in_gt, out = mask_pos (in place).
// ---------------------------------------------------------------------------
__global__ __launch_bounds__(256) void k_topk(const float* __restrict__ align_i,
                                              const float* __restrict__ gt_mask,
                                              float* maskbuf) {
  const int bm  = blockIdx.x;
  const int tid = threadIdx.x;
  const float gm = gt_mask[bm];
  const float* row = align_i + (size_t)bm * A_;
  float* mb = maskbuf + (size_t)bm * A_;

  if (gm <= 0.0f) {
    for (int a = tid; a < A_; a += 256) mb[a] = 0.0f;
    return;
  }

  __shared__ float sv[256];
  __shared__ int   si[256];
  __shared__ int   topIdx[TOPK_];
  __shared__ float lastV;
  __shared__ int   lastI;
  if (tid == 0) { lastV = 3.402823e38f; lastI = -1; }
  __syncthreads();

  for (int t = 0; t < TOPK_; ++t) {
    const float lv = lastV;
    const int   li = lastI;
    float bestV = -1.0f;
    int   bestI = 0x7fffffff;
    for (int a = tid; a < A_; a += 256) {
      const float v = row[a];
      const bool elig = (v < lv) || (v == lv && a > li);
      if (elig && (v > bestV || (v == bestV && a < bestI))) { bestV = v; bestI = a; }
    }
    sv[tid] = bestV; si[tid] = bestI;
    __syncthreads();
    for (int s = 128; s > 0; s >>= 1) {
      if (tid < s) {
        const float ov = sv[tid + s]; const int oi = si[tid + s];
        if (ov > sv[tid] || (ov == sv[tid] && oi < si[tid])) { sv[tid] = ov; si[tid] = oi; }
      }
      __syncthreads();
    }
    if (tid == 0) { topIdx[t] = si[0]; lastV = sv[0]; lastI = si[0]; }
    __syncthreads();
  }

  for (int a = tid; a < A_; a += 256) {
    float sel = 0.0f;
    #pragma unroll
    for (int t = 0; t < TOPK_; ++t) sel = (a == topIdx[t]) ? 1.0f : sel;
    mb[a] = sel * mb[a] * gm;     // mask_topk * mask_in_gt * gt_mask
  }
}

// ---------------------------------------------------------------------------
// Kernel 4: per (b,a) column over m: resolve multi-GT anchors via argmax-iou
// one-hot, emit target_labels / target_bboxes / tgt_idx / fg_mask.
// ---------------------------------------------------------------------------
__global__ __launch_bounds__(256) void k_col(const float* __restrict__ iou_i,
                                             float* mask_pos,
                                             const int*   __restrict__ labels,
                                             const float* __restrict__ gt_bboxes,
                                             float* __restrict__ out_labels,
                                             float* __restrict__ out_bbox,
                                             float* __restrict__ out_tgt,
                                             float* __restrict__ out_fg) {
  const int b = blockIdx.y;
  const int a = blockIdx.x * 256 + threadIdx.x;
  if (a >= A_) return;

  float fg = 0.0f, bestIou = -1.0f;
  int bestM = 0;
  for (int m = 0; m < M_; ++m) {
    const size_t off = ((size_t)(b * M_ + m)) * A_ + a;
    fg += mask_pos[off];
    const float iv = iou_i[off];
    if (iv > bestIou) { bestIou = iv; bestM = m; }   // first max (argmax)
  }

  int tgt;
  float fgOut;
  if (fg > 1.0f) {
    for (int m = 0; m < M_; ++m)
      mask_pos[((size_t)(b * M_ + m)) * A_ + a] = (m == bestM) ? 1.0f : 0.0f;
    tgt = bestM;
    fgOut = 1.0f;
  } else {
    tgt = 0;
    for (int m = 0; m < M_; ++m) {
      if (mask_pos[((size_t)(b * M_ + m)) * A_ + a] > 0.5f) { tgt = m; break; }
    }
    fgOut = (fg > 0.0f) ? 1.0f : 0.0f;
  }

  const size_t ba = (size_t)b * A_ + a;
  int lab = labels[b * M_ + tgt];
  lab = lab < 0 ? 0 : lab;
  out_labels[ba] = (float)lab;
  const float4 gb = ((const float4*)gt_bboxes)[b * M_ + tgt];
  out_bbox[ba * 4 + 0] = gb.x; out_bbox[ba * 4 + 1] = gb.y;
  out_bbox[ba * 4 + 2] = gb.z; out_bbox[ba * 4 + 3] = gb.w;
  out_tgt[ba] = (float)tgt;
  out_fg[ba]  = fgOut;
}

// ---------------------------------------------------------------------------
// Kernel 5: per (b,m): pos_align = max_a align*mask, pos_ov = max_a iou*mask.
// ---------------------------------------------------------------------------
__global__ __launch_bounds__(256) void k_posmax(const float* __restrict__ align_i,
                                                const float* __restrict__ iou_i,
                                                const float* __restrict__ mask_pos,
                                                float* __restrict__ pos_align,
                                                float* __restrict__ pos_ov) {
  const int bm = blockIdx.x;
  const size_t base = (size_t)bm * A_;
  float ma = 0.0f, mo = 0.0f;
  for (int a = threadIdx.x; a < A_; a += 256) {
    const float mp = mask_pos[base + a];
    ma = fmaxf(ma, align_i[base + a] * mp);
    mo = fmaxf(mo, iou_i[base + a] * mp);
  }
  __shared__ float sa[256], so[256];
  sa[threadIdx.x] = ma; so[threadIdx.x] = mo;
  __syncthreads();
  for (int s = 128; s > 0; s >>= 1) {
    if (threadIdx.x < s) {
      sa[threadIdx.x] = fmaxf(sa[threadIdx.x], sa[threadIdx.x + s]);
      so[threadIdx.x] = fmaxf(so[threadIdx.x], so[threadIdx.x + s]);
    }
    __syncthreads();
  }
  if (threadIdx.x == 0) { pos_align[bm] = sa[0]; pos_ov[bm] = so[0]; }
}

// ---------------------------------------------------------------------------
// Kernel 6: norm[b,a] = max_m align*mask * pos_ov / (pos_align + EPS).
// ---------------------------------------------------------------------------
__global__ __launch_bounds__(256) void k_norm(const float* __restrict__ align_i,
                                              const float* __restrict__ mask_pos,
                                              const float* __restrict__ pos_align,
                                              const float* __restrict__ pos_ov,
                                              float* __restrict__ normv) {
  const int b = blockIdx.y;
  const int a = blockIdx.x * 256 + threadIdx.x;
  if (a >= A_) return;
  float best = 0.0f;                     // all terms are >= 0
  for (int m = 0; m < M_; ++m) {
    const int bm = b * M_ + m;
    const size_t off = (size_t)bm * A_ + a;
    const float t = align_i[off] * mask_pos[off] * pos_ov[bm] / (pos_align[bm] + EPS_);
    best = fmaxf(best, t);
  }
  normv[(size_t)b * A_ + a] = best;
}

// ---------------------------------------------------------------------------
// Kernel 7: target_scores[b,a,c] = (fg && c==label) ? norm : 0. Coalesced.
// ---------------------------------------------------------------------------
__global__ __launch_bounds__(256) void k_scores(const float* __restrict__ normv,
                                                const float* __restrict__ out_labels,
                                                const float* __restrict__ out_fg,
                                                float* __restrict__ out_scores) {
  const size_t idx = (size_t)blockIdx.x * 256 + threadIdx.x;
  const size_t total = (size_t)B_ * A_ * NC_;
  if (idx >= total) return;
  const int c = (int)(idx % NC_);
  const size_t ba = idx / NC_;
  const float fg = out_fg[ba];
  const int lab = (int)out_labels[ba];
  out_scores[idx] = (fg > 0.0f && c == lab) ? normv[ba] : 0.0f;
}

// ---------------------------------------------------------------------------
extern "C" void kernel_launch(void* const* d_in, const int* in_sizes, int n_in,
                              void* d_out, int out_size, void* d_ws, size_t ws_size,
                              hipStream_t stream) {
  const float* pd_scores = (const float*)d_in[0];
  const float* pd_bboxes = (const float*)d_in[1];
  const float* anc       = (const float*)d_in[2];
  const int*   gt_labels = (const int*)  d_in[3];
  const float* gt_bboxes = (const float*)d_in[4];
  const float* gt_mask   = (const float*)d_in[5];
  float* out = (float*)d_out;

  const size_t BA   = (size_t)B_ * A_;       // 268800
  const size_t nBMA = (size_t)B_ * M_ * A_;  // 8601600

  // d_out layout (return-order concat):
  float* out_labels = out;                       // BA
  float* out_bbox   = out + BA;                  // BA*4
  float* out_scores = out + BA * 5;              // BA*NC
  float* out_tgt    = out + BA * 5 + BA * NC_;   // BA
  float* out_fg     = out_tgt + BA;              // BA

  // Workspace: arr0 = cls -> mask_in_gt -> mask_pos (reused in place).
  float* arr0  = (float*)d_ws;
  float* arr1  = arr0 + nBMA;        // align
  float* arr2  = arr1 + nBMA;        // iou
  float* posA  = arr2 + nBMA;        // B*M
  float* posO  = posA + B_ * M_;     // B*M
  float* normv = posO + B_ * M_;     // B*A

  dim3 gTile(A_ / 16, B_);
  dim3 gAnch((A_ + 255) / 256, B_);

  k_cls_wmma<<<gTile, 32, 0, stream>>>(pd_scores, gt_labels, arr0);
  k_pair<<<gAnch, 256, 0, stream>>>(pd_bboxes, anc, gt_bboxes, gt_mask,
                                    arr0, arr1, arr2);
  k_topk<<<dim3(B_ * M_), 256, 0, stream>>>(arr1, gt_mask, arr0);
  k_col<<<gAnch, 256, 0, stream>>>(arr2, arr0, gt_labels, gt_bboxes,
                                   out_labels, out_bbox, out_tgt, out_fg);
  k_posmax<<<dim3(B_ * M_), 256, 0, stream>>>(arr1, arr2, arr0, posA, posO);
  k_norm<<<gAnch, 256, 0, stream>>>(arr1, arr0, posA, posO, normv);

  const size_t nSc = BA * (size_t)NC_;
  k_scores<<<dim3((unsigned)((nSc + 255) / 256)), 256, 0, stream>>>(
      normv, out_labels, out_fg, out_scores);
}